// CLOULoss_90795608637707
// MI455X (gfx1250) — compile-verified
//
#include <hip/hip_runtime.h>

typedef __attribute__((ext_vector_type(2))) float v2f;
typedef __attribute__((ext_vector_type(8))) float v8f;

#define EPS_F 1e-6f

// ws layout (floats):
//   [0..63]      term1 block partials (64 blocks)
//   [64..191]    n_pred (128)
//   [192..319]   s_pred (128)
//   [320..383]   n_true (64)
//   [384..447]   s_true (64)
//   [512..16895] dist_pred (128*128)
//   [16896..20991] dt_cc (64*64)
// total 20992 floats = 82 KB

// ---------------- Kernel 1: per-row norm & sum ----------------
__global__ void clou_rowstats(const float* __restrict__ yp,
                              const float* __restrict__ yt,
                              float* __restrict__ n_pred, float* __restrict__ s_pred,
                              float* __restrict__ n_true, float* __restrict__ s_true) {
    int t = threadIdx.x;
    if (t < 128) {
        const float* r = yp + t * 64;
        float n = 0.f, s = 0.f;
        #pragma unroll 8
        for (int k = 0; k < 64; ++k) { float v = r[k]; n += v * v; s += v; }
        n_pred[t] = n; s_pred[t] = s;
    } else if (t < 192) {
        int i = t - 128;
        const float* r = yt + i * 64;
        float n = 0.f, s = 0.f;
        #pragma unroll 8
        for (int k = 0; k < 64; ++k) { float v = r[k]; n += v * v; s += v; }
        n_true[i] = n; s_true[i] = s;
    }
}

// ---------------- Kernel 2: Gram via fp32 WMMA -> distance matrix ----------------
// One wave (32 threads) computes one 16x16 tile of  D = Y * Y^T  (Y is N x 64),
// then converts to pairwise distance:
//   dist(i,j) = sqrt(n_i + n_j - 2*G_ij + 2*EPS*(s_j - s_i) + 64*EPS^2)
// fp32 WMMA layout (16x16x4):
//   A (16x4): lane m=L&15 holds row M=m; hi=L>>4 selects K pair {2hi, 2hi+1}
//   B (4x16): lane m holds col N=m; hi selects K pair {2hi, 2hi+1}
// -> both fragments are a contiguous float2 from row (tile*16 + m), col (4kk + 2hi).
__global__ __launch_bounds__(32) void clou_gram_dist(const float* __restrict__ Y,
                                                     const float* __restrict__ nrm,
                                                     const float* __restrict__ sum,
                                                     float* __restrict__ out, int N) {
    const int ti = blockIdx.y, tj = blockIdx.x;
    const int lane = threadIdx.x;
    const int m  = lane & 15;
    const int hi = lane >> 4;

    const float* Arow = Y + (ti * 16 + m) * 64;
    const float* Brow = Y + (tj * 16 + m) * 64;

    v8f acc = {};
    #pragma unroll
    for (int kk = 0; kk < 16; ++kk) {
        const int c = kk * 4 + hi * 2;
        v2f a = *(const v2f*)(Arow + c);
        v2f b = *(const v2f*)(Brow + c);
        acc = __builtin_amdgcn_wmma_f32_16x16x4_f32(
            /*neg_a=*/false, a, /*neg_b=*/false, b,
            /*c_mod=*/(short)0, acc, /*reuse_a=*/false, /*reuse_b=*/false);
    }

    const int j  = tj * 16 + m;          // N index = column = B row
    const float nj = nrm[j], sj = sum[j];
    #pragma unroll
    for (int r = 0; r < 8; ++r) {
        const int i = ti * 16 + r + hi * 8;   // M index = row = A row
        const float g  = acc[r];
        const float ni = nrm[i], si = sum[i];
        float d2 = ni + nj - 2.0f * g + 2.0f * EPS_F * (sj - si) + 64.0f * EPS_F * EPS_F;
        out[i * N + j] = sqrtf(fmaxf(d2, 0.0f));
    }
}

// ---------------- Kernel 3: term1 = sum softplus(p - t) over off-diagonal t ----------------
// 64 blocks x 256 threads; one thread per dist_pred element; dt_cc staged in LDS.
__global__ __launch_bounds__(256) void clou_softplus_reduce(const float* __restrict__ dp,
                                                            const float* __restrict__ dt,
                                                            float* __restrict__ partials) {
    __shared__ float t_lds[4096];
    __shared__ float red[256];
    const int tid = threadIdx.x;

    for (int i = tid; i < 4096; i += 256) t_lds[i] = dt[i];
    __syncthreads();

    const float p = dp[blockIdx.x * 256 + tid];
    float acc = 0.f;
    #pragma unroll 4
    for (int idx = 0; idx < 4096; ++idx) {
        if ((idx >> 6) == (idx & 63)) continue;     // skip diagonal (mask), uniform branch
        const float d = p - t_lds[idx];
        // numerically stable softplus: max(d,0) + log1p(exp(-|d|))
        acc += fmaxf(d, 0.f) + log1pf(expf(-fabsf(d)));
    }

    red[tid] = acc;
    __syncthreads();
    for (int s = 128; s > 0; s >>= 1) {
        if (tid < s) red[tid] += red[tid + s];
        __syncthreads();
    }
    if (tid == 0) partials[blockIdx.x] = red[0];
}

// ---------------- Kernel 4: finalize ----------------
__global__ __launch_bounds__(256) void clou_finalize(const float* __restrict__ partials,
                                                     const float* __restrict__ dt,
                                                     float* __restrict__ out) {
    __shared__ float red1[256];
    __shared__ float red2[256];
    const int tid = threadIdx.x;

    float t1 = (tid < 64) ? partials[tid] : 0.f;
    float t2 = 0.f;
    for (int idx = tid; idx < 4096; idx += 256) {
        if ((idx >> 6) != (idx & 63)) t2 += dt[idx];
    }
    red1[tid] = t1;
    red2[tid] = t2;
    __syncthreads();
    for (int s = 128; s > 0; s >>= 1) {
        if (tid < s) { red1[tid] += red1[tid + s]; red2[tid] += red2[tid + s]; }
        __syncthreads();
    }
    if (tid == 0) {
        // term2 = BETA * B*B * sum_offdiag ; denom = B * (C-1)^2 = 128 * 63^2
        const float loss = (red1[0] - 16384.0f * red2[0]) / 508032.0f;
        out[0] = loss;
    }
}

extern "C" void kernel_launch(void* const* d_in, const int* in_sizes, int n_in,
                              void* d_out, int out_size, void* d_ws, size_t ws_size,
                              hipStream_t stream) {
    const float* yp = (const float*)d_in[0];   // y_pred 128x64 f32
    const float* yt = (const float*)d_in[1];   // y_true 128x64 f32
    float* ws = (float*)d_ws;

    float* partials  = ws;           // 64
    float* n_pred    = ws + 64;      // 128
    float* s_pred    = ws + 192;     // 128
    float* n_true    = ws + 320;     // 64
    float* s_true    = ws + 384;     // 64
    float* dist_pred = ws + 512;     // 16384
    float* dt_cc     = ws + 16896;   // 4096

    clou_rowstats<<<1, 192, 0, stream>>>(yp, yt, n_pred, s_pred, n_true, s_true);
    clou_gram_dist<<<dim3(8, 8), 32, 0, stream>>>(yp, n_pred, s_pred, dist_pred, 128);
    clou_gram_dist<<<dim3(4, 4), 32, 0, stream>>>(yt, n_true, s_true, dt_cc, 64);
    clou_softplus_reduce<<<64, 256, 0, stream>>>(dist_pred, dt_cc, partials);
    clou_finalize<<<1, 256, 0, stream>>>(partials, dt_cc, (float*)d_out);
}